// DecodeYoloV2_22694607192621
// MI455X (gfx1250) — compile-verified
//
#include <hip/hip_runtime.h>
#include <math.h>

// ---------------- problem constants ----------------
#define NPRED      8000      // 5 anchors * 40 * 40
#define NPAD       8192
#define HW         1600      // 40*40
#define GRID_W     40
#define NCLASS     80
#define STRIDEF    32.0f     // 1280 / 40
#define CONF_THR   0.5f
#define IOU_THR    0.45f
#define MROW_WORDS 256       // padded mask row stride in 32-bit words
#define REAL_WORDS 250       // 8000 bits

typedef float v2f __attribute__((ext_vector_type(2)));
typedef float v8f __attribute__((ext_vector_type(8)));

__device__ __forceinline__ float sigm(float t) { return 1.0f / (1.0f + expf(-t)); }
__device__ __forceinline__ float uf(unsigned u) { return __uint_as_float(u); }

// ================= 1) decode =================
__global__ void __launch_bounds__(256) k_decode(
    const float* __restrict__ x, const float* __restrict__ anch,
    float* __restrict__ raw,
    float* __restrict__ bx1, float* __restrict__ by1,
    float* __restrict__ bx2, float* __restrict__ by2,
    float* __restrict__ bar, int* __restrict__ bcls)
{
    int n = blockIdx.x * 256 + threadIdx.x;
    if (n >= NPRED) return;
    int a  = n / HW;
    int hw = n - a * HW;
    int hi = hw / GRID_W;
    int wi = hw - hi * GRID_W;
    const float* p = x + (size_t)a * 85 * HW + hw;
    float tx = p[0 * HW], ty = p[1 * HW], tw = p[2 * HW], th = p[3 * HW], tc = p[4 * HW];
    float best = p[5 * HW]; int bc = 0;
    for (int c = 1; c < NCLASS; ++c) {
        float v = p[(5 + c) * HW];
        if (v > best) { best = v; bc = c; }   // strict > keeps first occurrence (jnp.argmax)
    }
    float cx = (sigm(tx) + (float)wi) * STRIDEF;
    float cy = (sigm(ty) + (float)hi) * STRIDEF;
    float bw = expf(tw) * anch[2 * a + 0] * STRIDEF;
    float bh = expf(th) * anch[2 * a + 1] * STRIDEF;
    float conf = sigm(tc);
    raw[n * 6 + 0] = cx; raw[n * 6 + 1] = cy; raw[n * 6 + 2] = bw;
    raw[n * 6 + 3] = bh; raw[n * 6 + 4] = conf; raw[n * 6 + 5] = (float)bc;
    float x1 = cx - 0.5f * bw, y1 = cy - 0.5f * bh;
    float x2 = cx + 0.5f * bw, y2 = cy + 0.5f * bh;
    bx1[n] = x1; by1[n] = y1; bx2[n] = x2; by2[n] = y2;
    bar[n] = fmaxf(x2 - x1, 0.f) * fmaxf(y2 - y1, 0.f);
    bcls[n] = bc;
}

// ================= 2) bitonic argsort (desc score, asc index) + gather =================
__global__ void __launch_bounds__(1024) k_sort(
    const float* __restrict__ raw,
    const float* __restrict__ bx1, const float* __restrict__ by1,
    const float* __restrict__ bx2, const float* __restrict__ by2,
    const float* __restrict__ bar, const int* __restrict__ bcls,
    float* __restrict__ sx1, float* __restrict__ sy1,
    float* __restrict__ sx2, float* __restrict__ sy2,
    float* __restrict__ sar, int* __restrict__ scls,
    int* __restrict__ svalid, int* __restrict__ order)
{
    __shared__ unsigned long long keys[NPAD];   // 64 KB of the 320 KB WGP LDS
    int tid = threadIdx.x;
    for (int i = tid; i < NPAD; i += 1024) {
        unsigned long long key;
        if (i < NPRED) {
            float conf = raw[i * 6 + 4];
            float s = conf > CONF_THR ? conf : -1.0f;       // reference: where(valid, conf, -1)
            unsigned u = __float_as_uint(s);
            u ^= (u & 0x80000000u) ? 0xFFFFFFFFu : 0x80000000u;  // order-preserving encode
            key = ((unsigned long long)(~u) << 32) | (unsigned)i; // desc score, asc index
        } else {
            key = ~0ull;                                     // pad sorts last
        }
        keys[i] = key;
    }
    __syncthreads();
    for (int k = 2; k <= NPAD; k <<= 1) {
        for (int j = k >> 1; j > 0; j >>= 1) {
            for (int i = tid; i < NPAD; i += 1024) {
                int ixj = i ^ j;
                if (ixj > i) {
                    bool up = ((i & k) == 0);
                    unsigned long long a = keys[i], b = keys[ixj];
                    if ((a > b) == up) { keys[i] = b; keys[ixj] = a; }
                }
            }
            __syncthreads();
        }
    }
    for (int i = tid; i < NPAD; i += 1024) {
        if (i < NPRED) {
            int n = (int)(unsigned)keys[i];
            order[i] = n;
            sx1[i] = bx1[n]; sy1[i] = by1[n]; sx2[i] = bx2[n]; sy2[i] = by2[n];
            sar[i] = bar[n]; scls[i] = bcls[n];
            svalid[i] = raw[n * 6 + 4] > CONF_THR ? 1 : 0;
        } else {
            order[i] = -1;
            sx1[i] = sy1[i] = sx2[i] = sy2[i] = sar[i] = 0.f;
            scls[i] = -2; svalid[i] = 0;    // degenerate box: iou == 0
        }
    }
}

// ================= 3) suppression bit-matrix: one wave per 16x32 tile =================
// Per-wave LDS slice layout (288 dwords):
//   [0..15]=rowX1 [16..31]=rowY1 [32..47]=rowX2 [48..63]=rowY2 [64..79]=rowArea [80..95]=rowCls
//   [96..127]=colX1 [128..159]=colY1 [160..191]=colX2 [192..223]=colY2 [224..255]=colArea [256..287]=colCls
__global__ void __launch_bounds__(256) k_mask(
    const float* __restrict__ sx1, const float* __restrict__ sy1,
    const float* __restrict__ sx2, const float* __restrict__ sy2,
    const float* __restrict__ sar, const int* __restrict__ scls,
    unsigned* __restrict__ mask)
{
    __shared__ unsigned smem[8][288];
    int wave = threadIdx.x >> 5;
    int lane = threadIdx.x & 31;
    int task = blockIdx.x * 8 + wave;        // 128000 tasks = 500 rowTiles * 256 word strips
    int rowTile = task >> 8;                 // 0..499
    int wordIdx = task & 255;                // 0..255
    int rowBase = rowTile * 16;
    int colBase = wordIdx * 32;
    unsigned* sm = &smem[wave][0];
    // LDS byte address: low 32 bits of the flat pointer are the LDS offset (aperture in [63:32])
    unsigned ldsBase = (unsigned)(unsigned long long)(void*)sm;

    // ---- async gather of 288 dwords into LDS (CDNA5 async-to-LDS path, ASYNCcnt) ----
    #pragma unroll
    for (int k = 0; k < 9; ++k) {
        int slot = k * 32 + lane;
        int arr, idx;
        if (slot < 96) { arr = slot >> 4;          idx = rowBase + (slot & 15); }
        else           { int c = slot - 96; arr = c >> 5; idx = colBase + (c & 31); }
        const void* p = (arr == 0) ? (const void*)sx1 :
                        (arr == 1) ? (const void*)sy1 :
                        (arr == 2) ? (const void*)sx2 :
                        (arr == 3) ? (const void*)sy2 :
                        (arr == 4) ? (const void*)sar : (const void*)scls;
        unsigned long long g = (unsigned long long)(size_t)p + (unsigned long long)idx * 4ull;
        unsigned l = ldsBase + (unsigned)slot * 4u;
        asm volatile("global_load_async_to_lds_b32 %0, %1, off"
                     :: "v"(l), "v"(g) : "memory");
    }
    asm volatile("s_wait_asynccnt 0" ::: "memory");

    // ---- WMMA: D[i][j] = area_i + area_j as a rank-2 outer product (16x16x4 f32) ----
    // A (16x4): lanes 0-15 hold M=lane, {K0,K1}; lanes 16-31 hold {K2,K3} (zeros here).
    // B (4x16): lanes 0-15 hold N=lane, {K0,K1}; lanes 16-31 hold {K2,K3} (zeros here).
    int n0 = lane & 15;
    v2f A, B0, B1;
    if (lane < 16) {
        A.x  = uf(sm[64 + lane]);  A.y  = 1.0f;                   // [area_i, 1]
        B0.x = 1.0f;               B0.y = uf(sm[224 + lane]);      // [1, area_j]  (cols 0..15)
        B1.x = 1.0f;               B1.y = uf(sm[224 + 16 + lane]); //              (cols 16..31)
    } else {
        A.x = 0.f; A.y = 0.f; B0.x = 0.f; B0.y = 0.f; B1.x = 0.f; B1.y = 0.f;
    }
    v8f c0 = {};
    v8f d0 = __builtin_amdgcn_wmma_f32_16x16x4_f32(false, A, false, B0, (short)0, c0, false, false);
    v8f d1 = __builtin_amdgcn_wmma_f32_16x16x4_f32(false, A, false, B1, (short)0, c0, false, false);

    // per-lane column boxes (fixed across the 8 accumulator rows)
    float c0x1 = uf(sm[96 + n0]),       c0y1 = uf(sm[128 + n0]);
    float c0x2 = uf(sm[160 + n0]),      c0y2 = uf(sm[192 + n0]);
    int   c0c  = (int)sm[256 + n0];
    float c1x1 = uf(sm[96 + 16 + n0]),  c1y1 = uf(sm[128 + 16 + n0]);
    float c1x2 = uf(sm[160 + 16 + n0]), c1y2 = uf(sm[192 + 16 + n0]);
    int   c1c  = (int)sm[256 + 16 + n0];

    int rsel = (lane >> 4) << 3;       // 0 for lanes 0-15, 8 for lanes 16-31
    unsigned words[16];
    #pragma unroll
    for (int v = 0; v < 8; ++v) {
        int r = v + rsel;                                  // C/D layout: VGPR v -> row v / v+8
        float rx1 = uf(sm[0 + r]),  ry1 = uf(sm[16 + r]);
        float rx2 = uf(sm[32 + r]), ry2 = uf(sm[48 + r]);
        int   rc  = (int)sm[80 + r];

        // denom = area_i + area_j - inter + eps  > 0 always, so
        //   inter/denom > T  <=>  inter > T*denom   (no division needed)
        float iw0 = fminf(rx2, c0x2) - fmaxf(rx1, c0x1);
        float ih0 = fminf(ry2, c0y2) - fmaxf(ry1, c0y1);
        float in0 = fmaxf(iw0, 0.f) * fmaxf(ih0, 0.f);
        float den0 = d0[v] - in0 + 1e-9f;
        unsigned m0 = __builtin_amdgcn_ballot_w32((in0 > IOU_THR * den0) && (rc == c0c));

        float iw1 = fminf(rx2, c1x2) - fmaxf(rx1, c1x1);
        float ih1 = fminf(ry2, c1y2) - fmaxf(ry1, c1y1);
        float in1 = fmaxf(iw1, 0.f) * fmaxf(ih1, 0.f);
        float den1 = d1[v] - in1 + 1e-9f;
        unsigned m1 = __builtin_amdgcn_ballot_w32((in1 > IOU_THR * den1) && (rc == c1c));

        // ballot: low 16 lanes -> row v cols 0-15, high 16 lanes -> row v+8 cols 0-15
        words[v]     = (m0 & 0xFFFFu) | (m1 << 16);
        words[v + 8] = (m0 >> 16)     | (m1 & 0xFFFF0000u);
    }
    if (lane == 0) {
        #pragma unroll
        for (int r = 0; r < 16; ++r)
            mask[(unsigned)(rowBase + r) * MROW_WORDS + wordIdx] = words[r];
    }
}

// ================= 4) sequential greedy scan using precomputed mask rows =================
__global__ void __launch_bounds__(1024) k_nms_seq(
    const unsigned* __restrict__ mask, const int* __restrict__ svalid,
    const int* __restrict__ order, const float* __restrict__ raw,
    float* __restrict__ out)
{
    __shared__ unsigned sup[MROW_WORDS];
    __shared__ unsigned keepBits[MROW_WORDS];
    int tid = threadIdx.x;
    for (int w = tid; w < MROW_WORDS; w += 1024) { sup[w] = 0u; keepBits[w] = 0u; }
    __syncthreads();
    for (int i = 0; i < NPRED; ++i) {
        bool active = (svalid[i] != 0) && (((sup[i >> 5] >> (i & 31)) & 1u) == 0u);
        __syncthreads();                       // everyone reads sup before anyone ORs
        if (active) {
            const unsigned* row = mask + (size_t)i * MROW_WORDS;
            for (int w = tid; w < REAL_WORDS; w += 1024) sup[w] |= row[w];
        }
        if (tid == 0 && active) keepBits[i >> 5] |= 1u << (i & 31);
        __syncthreads();
    }
    for (int i = tid; i < NPRED; i += 1024) {
        int n = order[i];
        float k = ((keepBits[i >> 5] >> (i & 31)) & 1u) ? 1.0f : 0.0f;
        #pragma unroll
        for (int c = 0; c < 6; ++c) out[n * 6 + c] = raw[n * 6 + c] * k;
        out[NPRED * 6 + n] = k;
    }
}

// ---- fallback: same scan but IoU computed on the fly (used if ws too small for the mask) ----
__global__ void __launch_bounds__(1024) k_nms_fallback(
    const float* __restrict__ sx1, const float* __restrict__ sy1,
    const float* __restrict__ sx2, const float* __restrict__ sy2,
    const float* __restrict__ sar, const int* __restrict__ scls,
    const int* __restrict__ svalid, const int* __restrict__ order,
    const float* __restrict__ raw, float* __restrict__ out)
{
    __shared__ unsigned sup[MROW_WORDS];
    __shared__ unsigned keepBits[MROW_WORDS];
    int tid = threadIdx.x;
    for (int w = tid; w < MROW_WORDS; w += 1024) { sup[w] = 0u; keepBits[w] = 0u; }
    __syncthreads();
    for (int i = 0; i < NPRED; ++i) {
        bool active = (svalid[i] != 0) && (((sup[i >> 5] >> (i & 31)) & 1u) == 0u);
        __syncthreads();
        if (active) {
            float x1 = sx1[i], y1 = sy1[i], x2 = sx2[i], y2 = sy2[i], ar = sar[i];
            int   c  = scls[i];
            for (int j = tid; j < NPRED; j += 1024) {
                float iw = fminf(x2, sx2[j]) - fmaxf(x1, sx1[j]);
                float ih = fminf(y2, sy2[j]) - fmaxf(y1, sy1[j]);
                float in = fmaxf(iw, 0.f) * fmaxf(ih, 0.f);
                float den = ar + sar[j] - in + 1e-9f;
                if ((in > IOU_THR * den) && scls[j] == c)
                    atomicOr(&sup[j >> 5], 1u << (j & 31));
            }
        }
        if (tid == 0 && active) keepBits[i >> 5] |= 1u << (i & 31);
        __syncthreads();
    }
    for (int i = tid; i < NPRED; i += 1024) {
        int n = order[i];
        float k = ((keepBits[i >> 5] >> (i & 31)) & 1u) ? 1.0f : 0.0f;
        #pragma unroll
        for (int c = 0; c < 6; ++c) out[n * 6 + c] = raw[n * 6 + c] * k;
        out[NPRED * 6 + n] = k;
    }
}

// ================= launch =================
extern "C" void kernel_launch(void* const* d_in, const int* in_sizes, int n_in,
                              void* d_out, int out_size, void* d_ws, size_t ws_size,
                              hipStream_t stream)
{
    const float* x    = (const float*)d_in[0];
    const float* anch = (const float*)d_in[1];
    float* out = (float*)d_out;
    char* ws = (char*)d_ws;
    size_t off = 0;
    auto carve = [&](size_t nbytes) -> void* {
        off = (off + 255) & ~(size_t)255;
        void* p = ws + off;
        off += nbytes;
        return p;
    };
    float* raw  = (float*)carve(NPRED * 6 * sizeof(float));
    float* bx1  = (float*)carve(NPRED * sizeof(float));
    float* by1  = (float*)carve(NPRED * sizeof(float));
    float* bx2  = (float*)carve(NPRED * sizeof(float));
    float* by2  = (float*)carve(NPRED * sizeof(float));
    float* bar  = (float*)carve(NPRED * sizeof(float));
    int*   bcls = (int*)carve(NPRED * sizeof(int));
    float* sx1  = (float*)carve(NPAD * sizeof(float));
    float* sy1  = (float*)carve(NPAD * sizeof(float));
    float* sx2  = (float*)carve(NPAD * sizeof(float));
    float* sy2  = (float*)carve(NPAD * sizeof(float));
    float* sar  = (float*)carve(NPAD * sizeof(float));
    int*   scls = (int*)carve(NPAD * sizeof(int));
    int*   svld = (int*)carve(NPAD * sizeof(int));
    int*   ordr = (int*)carve(NPAD * sizeof(int));
    size_t base_needed = off;
    unsigned* mask = (unsigned*)carve((size_t)NPRED * MROW_WORDS * sizeof(unsigned));
    size_t mask_needed = off;

    k_decode<<<dim3((NPRED + 255) / 256), dim3(256), 0, stream>>>(
        x, anch, raw, bx1, by1, bx2, by2, bar, bcls);
    k_sort<<<dim3(1), dim3(1024), 0, stream>>>(
        raw, bx1, by1, bx2, by2, bar, bcls,
        sx1, sy1, sx2, sy2, sar, scls, svld, ordr);

    if (ws_size >= mask_needed) {
        // 500 row-tiles * 256 word strips = 128000 wave tasks, 8 waves per block
        k_mask<<<dim3(16000), dim3(256), 0, stream>>>(sx1, sy1, sx2, sy2, sar, scls, mask);
        k_nms_seq<<<dim3(1), dim3(1024), 0, stream>>>(mask, svld, ordr, raw, out);
    } else if (ws_size >= base_needed) {
        k_nms_fallback<<<dim3(1), dim3(1024), 0, stream>>>(
            sx1, sy1, sx2, sy2, sar, scls, svld, ordr, raw, out);
    }
    (void)in_sizes; (void)n_in; (void)out_size;
}